// ConvCheb_41815801594275
// MI455X (gfx1250) — compile-verified
//
#include <hip/hip_runtime.h>

// ChebConv K=3 for MI455X (gfx1250, wave32).
// Phase 1: x1 = L @ x0            (COO scatter, native f32 atomics)
// Phase 2: x2 = 2*(L @ x1) - x0   (x2 pre-initialized to -x0, scaled scatter)
// Phase 3: out = [x0|x1|x2] @ W + bias  via V_WMMA_F32_16X16X4_F32
//
// Layouts:
//   x0 == inputs, [B][V][64] (read strided per-b)
//   x1, x2 in workspace as [V][B][64]  (row v contiguous 512 floats)
//   weight [Fin=64][K=3][Fout=64] -> staged to LDS K-pair-interleaved:
//     wlds[m][row/2][fo][row%2]  so a B-fragment is ONE aligned ds_load_b64.

typedef float v2f __attribute__((ext_vector_type(2)));
typedef float v8f __attribute__((ext_vector_type(8)));

#define FIN  64
#define FOUT 64
#define NB   8
#define COLS 512   // NB*FIN

// ---------------- Phase 0: x1 = 0, x2 = -x0 ----------------
__global__ void __launch_bounds__(256) cheb_init(
    const float* __restrict__ inputs, float* __restrict__ x1,
    float* __restrict__ x2, int V) {
  size_t i = (size_t)blockIdx.x * 256 + threadIdx.x;  // float4 units: V*128 total
  int v   = (int)(i >> 7);
  int rem = (int)(i & 127);
  int b   = rem >> 4;
  int fi0 = (rem & 15) << 2;
  const float4 x = *(const float4*)(inputs + ((size_t)b * V + v) * FIN + fi0);
  float4 z;  z.x = 0.f;   z.y = 0.f;   z.z = 0.f;   z.w = 0.f;
  float4 nx; nx.x = -x.x; nx.y = -x.y; nx.z = -x.z; nx.w = -x.w;
  ((float4*)x1)[i] = z;    // [v][b][fi] flat == i*4
  ((float4*)x2)[i] = nx;
}

// ---------------- Phase 1: x1 += val * x0[col] ----------------
// One block per edge: rows/cols/vals are wave-uniform (scalar loads).
__global__ void __launch_bounds__(128) cheb_spmm_x1(
    const float* __restrict__ inputs, const int* __restrict__ rows,
    const int* __restrict__ cols, const float* __restrict__ vals,
    float* __restrict__ x1, int V) {
  int e = blockIdx.x;
  int r = rows[e];
  int c = cols[e];
  float val = vals[e];
  int t   = threadIdx.x;        // 0..127 covers 512 floats
  int b   = t >> 4;
  int fi0 = (t & 15) << 2;
  const float4 s = *(const float4*)(inputs + ((size_t)b * V + c) * FIN + fi0);
  float* dst = x1 + (((size_t)r * NB + b) * FIN + fi0);
  unsafeAtomicAdd(dst + 0, val * s.x);
  unsafeAtomicAdd(dst + 1, val * s.y);
  unsafeAtomicAdd(dst + 2, val * s.z);
  unsafeAtomicAdd(dst + 3, val * s.w);
}

// ---------------- Phase 2: x2 += 2*val * x1[col] ----------------
__global__ void __launch_bounds__(128) cheb_spmm_x2(
    const float* __restrict__ x1, const int* __restrict__ rows,
    const int* __restrict__ cols, const float* __restrict__ vals,
    float* __restrict__ x2, int V) {
  int e = blockIdx.x;
  int r = rows[e];
  int c = cols[e];
  float val = 2.0f * vals[e];
  int t   = threadIdx.x;
  int b   = t >> 4;
  int fi0 = (t & 15) << 2;
  const float4 s = *(const float4*)(x1 + (((size_t)c * NB + b) * FIN + fi0));
  float* dst = x2 + (((size_t)r * NB + b) * FIN + fi0);
  unsafeAtomicAdd(dst + 0, val * s.x);
  unsafeAtomicAdd(dst + 1, val * s.y);
  unsafeAtomicAdd(dst + 2, val * s.z);
  unsafeAtomicAdd(dst + 3, val * s.w);
}

// ---------------- Phase 3: WMMA GEMM ----------------
// One wave: 16 output rows x 64 output cols. K loop = 3 matrices x 16 steps of K=4.
// A-frag (16x4 f32): lanes 0-15 -> M=lane, VGPR{0,1}=K{0,1}; lanes 16-31 -> K{2,3}.
// B-frag (4x16 f32): mirrored K split, N = lane&15.  One ds_load_b64 per frag.
// C/D (16x16 f32): VGPR r -> M = r + (lane<16?0:8), N = lane&15.
__device__ __forceinline__ void cheb_kslab(const float* __restrict__ ap,
                                           const float* wbase, int koff,
                                           int lane15, v8f acc[4]) {
#pragma unroll
  for (int kk = 0; kk < 16; ++kk) {
    int fi = kk << 2;
    v2f a = *(const v2f*)(ap + fi + koff);
    // pair-interleaved: rows (fi+koff, fi+koff+1) at column n are adjacent
    const float* wl = wbase + ((fi + koff) >> 1) * (2 * FOUT) + lane15 * 2;
#pragma unroll
    for (int nt = 0; nt < 4; ++nt) {
      v2f bf = *(const v2f*)(wl + nt * 32);   // column nt*16+lane15
      acc[nt] = __builtin_amdgcn_wmma_f32_16x16x4_f32(
          false, a, false, bf, (short)0, acc[nt], false, false);
    }
  }
}

__global__ void __launch_bounds__(256) cheb_gemm(
    const float* __restrict__ x0,      // inputs [B][V][64]
    const float* __restrict__ x1,      // [V][8][64]
    const float* __restrict__ x2,      // [V][8][64]
    const float* __restrict__ weight,  // [64][3][64]
    const float* __restrict__ bias,    // [64]
    float* __restrict__ out, int V) {
  // 48 KB: wlds[m][row/2][fo][row%2]
  __shared__ float wlds[3 * FIN * FOUT];
  for (int d = threadIdx.x; d < 3 * FIN * FOUT; d += 256) {
    int m    = d >> 12;
    int rem  = d & 4095;
    int fi2  = rem >> 7;          // row pair
    int rest = rem & 127;
    int fo   = rest >> 1;
    int lo   = rest & 1;
    int row  = fi2 * 2 + lo;
    wlds[d] = weight[row * (3 * FOUT) + m * FOUT + fo];
  }
  __syncthreads();

  int wave = threadIdx.x >> 5;
  int lane = threadIdx.x & 31;
  int tile = blockIdx.x * 8 + wave;
  int rowbase = tile * 16;          // global output row (b*V + v)
  int b  = rowbase / V;             // V % 16 == 0: tile never crosses b
  int v0 = rowbase - b * V;
  int mrow   = lane & 15;
  int koff   = (lane < 16) ? 0 : 2;
  int lane15 = lane & 15;

  const float* aptr0 = x0 + ((size_t)b * V + v0 + mrow) * FIN;
  const float* aptr1 = x1 + (((size_t)(v0 + mrow)) * NB + b) * FIN;
  const float* aptr2 = x2 + (((size_t)(v0 + mrow)) * NB + b) * FIN;

  v8f acc[4] = {};
  cheb_kslab(aptr0, wlds,                  koff, lane15, acc);
  cheb_kslab(aptr1, wlds + 1 * FIN * FOUT, koff, lane15, acc);
  cheb_kslab(aptr2, wlds + 2 * FIN * FOUT, koff, lane15, acc);

  int Mofs = (lane < 16) ? 0 : 8;
#pragma unroll
  for (int nt = 0; nt < 4; ++nt) {
    float bv = bias[nt * 16 + lane15];
#pragma unroll
    for (int r = 0; r < 8; ++r) {
      out[((size_t)rowbase + Mofs + r) * FOUT + nt * 16 + lane15] = acc[nt][r] + bv;
    }
  }
}

extern "C" void kernel_launch(void* const* d_in, const int* in_sizes, int n_in,
                              void* d_out, int out_size, void* d_ws, size_t ws_size,
                              hipStream_t stream) {
  (void)n_in; (void)out_size; (void)ws_size;
  const float* inputs   = (const float*)d_in[0];
  const float* weight   = (const float*)d_in[1];
  const float* bias     = (const float*)d_in[2];
  const int*   lap_rows = (const int*)d_in[3];
  const int*   lap_cols = (const int*)d_in[4];
  const float* lap_vals = (const float*)d_in[5];

  const int V   = in_sizes[0] / (NB * FIN);   // 49152
  const int nnz = in_sizes[3];                // V*8

  float* x1 = (float*)d_ws;                   // V*512 floats
  float* x2 = x1 + (size_t)V * COLS;          // V*512 floats

  int initBlocks = (V * (COLS / 4)) / 256;    // float4 units / 256
  cheb_init<<<initBlocks, 256, 0, stream>>>(inputs, x1, x2, V);
  cheb_spmm_x1<<<nnz, 128, 0, stream>>>(inputs, lap_rows, lap_cols, lap_vals, x1, V);
  cheb_spmm_x2<<<nnz, 128, 0, stream>>>(x1, lap_rows, lap_cols, lap_vals, x2, V);

  int tiles = (NB * V) / 16;                  // 24576 waves of work
  cheb_gemm<<<tiles / 8, 256, 0, stream>>>(inputs, x1, x2, weight, bias,
                                           (float*)d_out, V);
}